// DetectionIntentionLoss_13855564496958
// MI455X (gfx1250) — compile-verified
//
#include <hip/hip_runtime.h>
#include <stdint.h>
#include <math.h>

// Problem constants (match reference)
#define A_N 131072
#define B_N 8
#define G_N 64
#define CINT 8

static constexpr float IOU_THR = 0.6f;
static constexpr float NEG_THR = 0.45f;
static constexpr float ALPHA_C = 0.25f;
static constexpr float BETA_C  = 1.0f / 9.0f;
static constexpr float EPS_C   = 1e-6f;

typedef float v2f __attribute__((ext_vector_type(2)));
typedef float v8f __attribute__((ext_vector_type(8)));

// Exact fp32 full-wave (32 lane) sum using the matrix pipe.
// A is 16x4 f32 (2 VGPRs/lane), B = ones (layout-insensitive):
// D[m][n] = sum_k A[m][k]; summing one full column of D (8 VGPRs + the
// opposite 16-lane half via shfl_xor(16)) gives the sum of ALL 64 A elements,
// i.e. sum over lanes of (a0 + a1). We put (x, 0) per lane -> wave sum of x.
// V_WMMA_F32_16X16X4_F32 is exact fp32 (RNE), so no precision is lost.
__device__ __forceinline__ float wave_sum_wmma(float x) {
  v2f a;  a[0] = x;   a[1] = 0.0f;
  v2f on; on[0] = 1.0f; on[1] = 1.0f;
  v8f c = {};
  c = __builtin_amdgcn_wmma_f32_16x16x4_f32(
      /*neg_a=*/false, a, /*neg_b=*/false, on,
      /*c_mod=*/(short)0, c, /*reuse_a=*/false, /*reuse_b=*/false);
  float s = c[0] + c[1] + c[2] + c[3] + c[4] + c[5] + c[6] + c[7];
  s += __shfl_xor(s, 16, 32);   // combine the two 16-lane halves (wave32)
  return s;                     // every lane holds the full wave sum
}

// ---------------------------------------------------------------- init
__global__ void k_init(unsigned long long* __restrict__ g_best) {
  int i = blockIdx.x * blockDim.x + threadIdx.x;
  if (i < B_N * G_N) g_best[i] = 0ull;
}

// ---------------------------------------------------------------- match
// Per (b,a): IoU over 64 GTs (from LDS), per-anchor max/argmax (first-index
// tie-break via strict >), and per-GT best-anchor via packed-key atomicMax.
__global__ __launch_bounds__(256) void k_match(
    const float* __restrict__ anchors, const float* __restrict__ gt_boxes,
    const int* __restrict__ gt_valid,
    float* __restrict__ max_iou_out, int* __restrict__ gt_idx_out,
    unsigned long long* __restrict__ g_best) {
  __shared__ float sx1[G_N], sx2[G_N], sy1[G_N], sy2[G_N], sar[G_N];
  __shared__ int   sval[G_N];
  __shared__ unsigned long long sbest[G_N];

  const int b = blockIdx.y;
  const int tid = threadIdx.x;
  if (tid < G_N) {
    const float* g = gt_boxes + (size_t)(b * G_N + tid) * 5;
    float cx = g[0], cy = g[1], w = g[2], l = g[3];
    sx1[tid] = cx - 0.5f * w;  sx2[tid] = cx + 0.5f * w;
    sy1[tid] = cy - 0.5f * l;  sy2[tid] = cy + 0.5f * l;
    sar[tid] = w * l;
    sval[tid] = gt_valid[b * G_N + tid];
    sbest[tid] = 0ull;
  }
  __syncthreads();

  const int a = blockIdx.x * 256 + tid;
  const float* an = anchors + (size_t)a * 5;
  float acx = an[0], acy = an[1], aw = an[2], al = an[3];
  float ax1 = acx - 0.5f * aw, ax2 = acx + 0.5f * aw;
  float ay1 = acy - 0.5f * al, ay2 = acy + 0.5f * al;
  float area_a = aw * al;

  float best = -2.0f;  int bidx = 0;
  for (int g = 0; g < G_N; ++g) {
    float iw = fmaxf(fminf(ax2, sx2[g]) - fmaxf(ax1, sx1[g]), 0.0f);
    float ih = fmaxf(fminf(ay2, sy2[g]) - fmaxf(ay1, sy1[g]), 0.0f);
    float inter = iw * ih;
    float iou = inter / (area_a + sar[g] - inter + EPS_C);
    iou = (sval[g] > 0) ? iou : -1.0f;
    if (iou > best) { best = iou; bidx = g; }   // strict > => first index wins
    // Only iou >= NEG_THR can ever satisfy force-match cond -> filter atomics.
    if (iou >= NEG_THR) {
      unsigned int fb = __float_as_uint(iou);
      unsigned int m  = fb ^ (((int)fb) < 0 ? 0xFFFFFFFFu : 0x80000000u);
      unsigned long long key =
          ((unsigned long long)m << 32) |
          (unsigned long long)(0xFFFFFFFFu - (unsigned)a);  // first anchor wins ties
      atomicMax(&sbest[g], key);
    }
  }
  max_iou_out[(size_t)b * A_N + a] = best;
  gt_idx_out[(size_t)b * A_N + a] = bidx;

  __syncthreads();
  if (tid < G_N && sbest[tid] != 0ull)
    atomicMax(&g_best[b * G_N + tid], sbest[tid]);
}

// ---------------------------------------------------------------- loss
__global__ __launch_bounds__(256) void k_loss(
    const float* __restrict__ cls_logits, const float* __restrict__ box_preds,
    const float* __restrict__ int_logits, const float* __restrict__ anchors,
    const float* __restrict__ gt_boxes, const int* __restrict__ gt_int,
    const float* __restrict__ max_iou_in, const int* __restrict__ gt_idx_in,
    const unsigned long long* __restrict__ g_best,
    float* __restrict__ partials) {
  __shared__ float sg[G_N][5];
  __shared__ int   sint[G_N];
  __shared__ int   sfa[G_N];       // forced anchor per GT (-1 if cond false)
  __shared__ float swave[8][4];

  const int b = blockIdx.y;
  const int tid = threadIdx.x;
  if (tid < G_N) {
    const float* g = gt_boxes + (size_t)(b * G_N + tid) * 5;
    sg[tid][0] = g[0]; sg[tid][1] = g[1]; sg[tid][2] = g[2];
    sg[tid][3] = g[3]; sg[tid][4] = g[4];
    sint[tid] = gt_int[b * G_N + tid];
    unsigned long long k = g_best[b * G_N + tid];
    // key written only when iou>=NEG_THR && valid -> key!=0 <=> cond true
    sfa[tid] = (k != 0ull)
                   ? (int)(0xFFFFFFFFu - (unsigned)(k & 0xFFFFFFFFull))
                   : -1;
  }
  __syncthreads();

  const int a = blockIdx.x * 256 + tid;
  const size_t ba = (size_t)b * A_N + a;
  float mi = max_iou_in[ba];
  int gi = gt_idx_in[ba];

  // forced = max g whose best anchor is `a` (ascending scan = .max() semantics)
  int forced = -1;
  for (int g = 0; g < G_N; ++g)
    if (sfa[g] == a) forced = g;

  bool pos = (mi >= IOU_THR) || (forced >= 0);
  bool neg = (mi < NEG_THR);
  if (forced >= 0) gi = forced;

  // ---- focal (over valid = pos|neg anchors)
  float x = cls_logits[ba];
  float t = pos ? 1.0f : 0.0f;
  float lg = log1pf(expf(-fabsf(x)));
  float lsp = fminf(x, 0.0f) - lg;    // log sigmoid(x)
  float lsn = fminf(-x, 0.0f) - lg;   // log sigmoid(-x)
  float ce = -(t * lsp + (1.0f - t) * lsn);
  float p = 1.0f / (1.0f + expf(-x));
  float p_t = p * t + (1.0f - p) * (1.0f - t);
  float a_t = ALPHA_C * t + (1.0f - ALPHA_C) * (1.0f - t);
  float om = 1.0f - p_t;
  float fsum = ((pos || neg) ? 1.0f : 0.0f) * a_t * ce * om * om;

  // ---- smooth-L1 + intention CE (positives only; masked in reference)
  float bsum = 0.0f, isum = 0.0f, psum = pos ? 1.0f : 0.0f;
  if (pos) {
    const float* an = anchors + (size_t)a * 5;
    float aw = an[2] + EPS_C, al = an[3] + EPS_C;
    float hd = sg[gi][4] - an[4];
    float tgt[6];
    tgt[0] = (sg[gi][0] - an[0]) / aw;
    tgt[1] = (sg[gi][1] - an[1]) / al;
    tgt[2] = logf(sg[gi][2] / aw + EPS_C);
    tgt[3] = logf(sg[gi][3] / al + EPS_C);
    tgt[4] = sinf(hd);
    tgt[5] = cosf(hd);
    const float* bp = box_preds + ba * 6;
    #pragma unroll
    for (int k = 0; k < 6; ++k) {
      float d = fabsf(bp[k] - tgt[k]);
      bsum += (d < BETA_C) ? 0.5f * d * d / BETA_C : d - 0.5f * BETA_C;
    }
    const float* il = int_logits + ba * CINT;
    float lv[CINT]; float mx = -INFINITY;
    #pragma unroll
    for (int k = 0; k < CINT; ++k) { lv[k] = il[k]; mx = fmaxf(mx, lv[k]); }
    float se = 0.0f;
    #pragma unroll
    for (int k = 0; k < CINT; ++k) se += expf(lv[k] - mx);
    isum = (mx + logf(se)) - lv[sint[gi]];
  }

  // ---- exact fp32 wave reduction on the matrix pipe (EXEC all-ones here)
  float wf = wave_sum_wmma(fsum);
  float wb = wave_sum_wmma(bsum);
  float wi = wave_sum_wmma(isum);
  float wp = wave_sum_wmma(psum);
  int wave = tid >> 5, lane = tid & 31;
  if (lane == 0) {
    swave[wave][0] = wf; swave[wave][1] = wb;
    swave[wave][2] = wi; swave[wave][3] = wp;
  }
  __syncthreads();
  if (tid == 0) {
    float s0 = 0.f, s1 = 0.f, s2 = 0.f, s3 = 0.f;
    for (int w = 0; w < 8; ++w) {
      s0 += swave[w][0]; s1 += swave[w][1];
      s2 += swave[w][2]; s3 += swave[w][3];
    }
    int blk = blockIdx.y * gridDim.x + blockIdx.x;
    partials[blk * 4 + 0] = s0;
    partials[blk * 4 + 1] = s1;
    partials[blk * 4 + 2] = s2;
    partials[blk * 4 + 3] = s3;
  }
}

// ---------------------------------------------------------------- finalize
__global__ __launch_bounds__(256) void k_final(
    const float* __restrict__ partials, int nblk, float* __restrict__ out) {
  __shared__ float acc[256][4];
  int tid = threadIdx.x;
  float s[4] = {0.f, 0.f, 0.f, 0.f};
  for (int i = tid; i < nblk; i += 256) {
    s[0] += partials[i * 4 + 0];
    s[1] += partials[i * 4 + 1];
    s[2] += partials[i * 4 + 2];
    s[3] += partials[i * 4 + 3];
  }
  for (int q = 0; q < 4; ++q) acc[tid][q] = s[q];
  __syncthreads();
  for (int st = 128; st > 0; st >>= 1) {
    if (tid < st)
      for (int q = 0; q < 4; ++q) acc[tid][q] += acc[tid + st][q];
    __syncthreads();
  }
  if (tid == 0) {
    float sum_f = acc[0][0], sum_b = acc[0][1], sum_i = acc[0][2], sum_p = acc[0][3];
    float np = fmaxf(sum_p, 1.0f);
    float lc = sum_f / np, lb = sum_b / np, li = sum_i / np;
    out[0] = lc + lb + 0.5f * li;  // CLS_W=1, BOX_W=1, INT_W=0.5
    out[1] = lc;
    out[2] = lb;
    out[3] = li;
    out[4] = sum_p;
  }
}

// ---------------------------------------------------------------- launch
extern "C" void kernel_launch(void* const* d_in, const int* in_sizes, int n_in,
                              void* d_out, int out_size, void* d_ws, size_t ws_size,
                              hipStream_t stream) {
  (void)in_sizes; (void)n_in; (void)out_size; (void)ws_size;
  const float* cls = (const float*)d_in[0];   // [B,A]
  const float* box = (const float*)d_in[1];   // [B,A,6]
  const float* itl = (const float*)d_in[2];   // [B,A,8]
  const float* anc = (const float*)d_in[3];   // [A,5]
  const float* gtb = (const float*)d_in[4];   // [B,G,5]
  const int*   gti = (const int*)d_in[5];     // [B,G]
  const int*   gtv = (const int*)d_in[6];     // [B,G]

  char* ws = (char*)d_ws;
  float* max_iou = (float*)ws;                                   // B*A f32 (4 MB)
  int*   gt_idx  = (int*)(ws + sizeof(float) * (size_t)B_N * A_N);   // 4 MB
  unsigned long long* g_best =
      (unsigned long long*)(ws + 2ull * sizeof(float) * B_N * A_N);  // 4 KB
  float* partials =
      (float*)(ws + 2ull * sizeof(float) * B_N * A_N +
               sizeof(unsigned long long) * (size_t)B_N * G_N);      // 64 KB

  dim3 grid(A_N / 256, B_N);
  k_init<<<2, 256, 0, stream>>>(g_best);
  k_match<<<grid, 256, 0, stream>>>(anc, gtb, gtv, max_iou, gt_idx, g_best);
  k_loss<<<grid, 256, 0, stream>>>(cls, box, itl, anc, gtb, gti,
                                   max_iou, gt_idx, g_best, partials);
  k_final<<<1, 256, 0, stream>>>(partials, (A_N / 256) * B_N, (float*)d_out);
}